// BPFTLoss_86406152061204
// MI455X (gfx1250) — compile-verified
//
#include <hip/hip_runtime.h>
#include <stdint.h>

#define TPB 256
#define NWAVES (TPB / 32)

typedef __attribute__((address_space(3))) float lds_f32;

namespace {

constexpr int   kB      = 2;
constexpr int   kS      = 2048;
constexpr int   kV      = 32000;         // floats per row
constexpr int   kSm1    = kS - 1;        // 2047
constexpr int   kRows   = kB * kSm1;     // 4094
constexpr float kLambda = 0.1f;

constexpr int kN4 = kV / 4;              // 8000 float4 per row
constexpr int kA4 = 4096;                // chunk A: 16 *full-EXEC* iterations of TPB
                                         // -> every wave issues exactly 16 async ops
// chunk B = kN4 - kA4 = 3904 -> 15..16 async ops per wave (>= 15 for all waves),
// so "s_wait_asynccnt 15" retires at least the first 16 (all of A) per wave.

__device__ __forceinline__ float wave_max(float v) {
#pragma unroll
    for (int m = 16; m > 0; m >>= 1) v = fmaxf(v, __shfl_xor(v, m, 32));
    return v;
}
__device__ __forceinline__ float wave_sum(float v) {
#pragma unroll
    for (int m = 16; m > 0; m >>= 1) v += __shfl_xor(v, m, 32);
    return v;
}

__global__ void __launch_bounds__(TPB)
bpft_row_kernel(const float* __restrict__ logits,
                const int*   __restrict__ labels,
                const float* __restrict__ fact,
                const float* __restrict__ contra,
                float*       __restrict__ row_part) {
    extern __shared__ float smem[];
    float* tile = smem;            // kV floats: one full vocab row
    float* scr  = smem + kV;       // 32 floats: cross-wave reduction scratch

    const int row = blockIdx.x;              // 0 .. kRows-1
    const int b   = row / kSm1;
    const int s   = row - b * kSm1;          // shifted-logits position
    const float* __restrict__ x = logits + ((size_t)b * kS + s) * (size_t)kV;

    const int wid = threadIdx.x >> 5;
    const int lid = threadIdx.x & 31;

    // ---- Stage the 125KB row into LDS with CDNA5 async global->LDS DMA ----
    const uint32_t lds_base = (uint32_t)(uintptr_t)(lds_f32*)tile;  // AS3 byte offset
    auto issue = [&](int i) {
        uint64_t ga = (uint64_t)(uintptr_t)(x + 4 * i);
        uint32_t la = lds_base + 16u * (uint32_t)i;
        asm volatile("global_load_async_to_lds_b128 %0, %1, off"
                     :: "v"(la), "v"(ga) : "memory");
    };
    for (int i = threadIdx.x; i < kA4; i += TPB) issue(i);        // chunk A (uniform)
    for (int i = kA4 + threadIdx.x; i < kN4; i += TPB) issue(i);  // chunk B

    const float f = fact[b];
    const float c = 0.1f * contra[b] * (1.0f - f);   // per-batch penalty scalar

    const float4* t4 = reinterpret_cast<const float4*>(tile);

    // ---- Pass 1: row maxima of x and y = x - c (overlapped with chunk-B DMA) ----
    float mx = -INFINITY, my = -INFINITY;
    auto max_range = [&](int lo, int hi) {
        for (int i = lo + (int)threadIdx.x; i < hi; i += TPB) {
            float4 v = t4[i];
            mx = fmaxf(mx, fmaxf(fmaxf(v.x, v.y), fmaxf(v.z, v.w)));
            float ya = v.x - c, yb = v.y - c, yc = v.z - c, yd = v.w - c;
            my = fmaxf(my, fmaxf(fmaxf(ya, yb), fmaxf(yc, yd)));
        }
    };
    asm volatile("s_wait_asynccnt 0xf" ::: "memory");   // wave's first 16 ops (chunk A) done
    __syncthreads();                                    // -> all of chunk A visible in LDS
    max_range(0, kA4);                                  // compute while chunk B streams in
    asm volatile("s_wait_asynccnt 0x0" ::: "memory");
    __syncthreads();                                    // -> full row in LDS
    max_range(kA4, kN4);

    // cross-wave max combine (one barrier pair, shared by mx/my)
    mx = wave_max(mx);
    my = wave_max(my);
    if (lid == 0) { scr[wid] = mx; scr[NWAVES + wid] = my; }
    __syncthreads();
    {
        float tmx = scr[0], tmy = scr[NWAVES];
#pragma unroll
        for (int w = 1; w < NWAVES; ++w) {
            tmx = fmaxf(tmx, scr[w]);
            tmy = fmaxf(tmy, scr[NWAVES + w]);
        }
        mx = tmx; my = tmy;
    }
    __syncthreads();

    // ---- Pass 2: Zx = sum e^(x-mx), Zy = sum e^(y-my), T = sum e^(y-my)*(y-x) ----
    float zx = 0.f, zy = 0.f, ts = 0.f;
    auto acc = [&](float xx) {
        float yy = xx - c;
        zx += __expf(xx - mx);
        float ey = __expf(yy - my);
        zy += ey;
        ts += ey * (yy - xx);
    };
    for (int i = threadIdx.x; i < kN4; i += TPB) {
        float4 v = t4[i];
        acc(v.x); acc(v.y); acc(v.z); acc(v.w);
    }
    zx = wave_sum(zx);
    zy = wave_sum(zy);
    ts = wave_sum(ts);
    if (lid == 0) { scr[wid] = zx; scr[NWAVES + wid] = zy; scr[2 * NWAVES + wid] = ts; }
    __syncthreads();

    if (threadIdx.x == 0) {
        float tzx = 0.f, tzy = 0.f, tts = 0.f;
#pragma unroll
        for (int w = 0; w < NWAVES; ++w) {
            tzx += scr[w];
            tzy += scr[NWAVES + w];
            tts += scr[2 * NWAVES + w];
        }
        const int   lab = labels[(size_t)b * kS + (s + 1)];   // shift_labels
        const float xl  = tile[lab];
        const float lzx = __logf(tzx);
        const float lzy = __logf(tzy);
        const float ce  = -(xl - mx - lzx);                   // -log_p[label]
        // KL row: sum_v q*(log q - log p), q = softmax(x - c)
        const float kl  = tts / tzy + (mx - my) + (lzx - lzy);
        row_part[row] = ce * (2.0f - f) * (1.0f / (float)kRows)
                      + kLambda * kl * (1.0f / (float)kB);
    }
}

__global__ void __launch_bounds__(TPB)
bpft_final_reduce(const float* __restrict__ row_part, float* __restrict__ out) {
    __shared__ float scr[NWAVES];
    float acc = 0.f;
    for (int i = threadIdx.x; i < kRows; i += TPB) acc += row_part[i];
    acc = wave_sum(acc);
    if ((threadIdx.x & 31) == 0) scr[threadIdx.x >> 5] = acc;
    __syncthreads();
    if (threadIdx.x == 0) {
        float total = 0.f;
#pragma unroll
        for (int w = 0; w < NWAVES; ++w) total += scr[w];
        out[0] = total;                 // overwrites poisoned d_out
    }
}

} // namespace

extern "C" void kernel_launch(void* const* d_in, const int* in_sizes, int n_in,
                              void* d_out, int out_size, void* d_ws, size_t ws_size,
                              hipStream_t stream) {
    const float* logits = (const float*)d_in[0];
    const int*   labels = (const int*)  d_in[1];   // harness ABI: integer -> const int*
    const float* fact   = (const float*)d_in[2];
    const float* contra = (const float*)d_in[3];

    float* row_part = (float*)d_ws;                // kRows floats of scratch

    const size_t shmem = (size_t)(kV + 32) * sizeof(float);    // ~128.1 KB / WG
    bpft_row_kernel<<<kRows, TPB, shmem, stream>>>(logits, labels, fact, contra, row_part);
    bpft_final_reduce<<<1, TPB, 0, stream>>>(row_part, (float*)d_out);
}